// HANLayer_69793218560332
// MI455X (gfx1250) — compile-verified
//
#include <hip/hip_runtime.h>
#include <hip/hip_bf16.h>

#define NN 50000
#define INS 256
#define HH 4
#define DD 64
#define HD 256      // H*D
#define EE 800000
#define NEG 0.2f

typedef __attribute__((ext_vector_type(2))) float v2f;
typedef __attribute__((ext_vector_type(8))) float v8f;

// ---------------------------------------------------------------------------
// fp32 WMMA GEMM:  C[M,Ncols] = A[M,K] * B[Ncols,K]^T (+ bias)
// One wave computes a 16x64 output stripe (A fragment reused across 4 column
// tiles / 4 independent accumulators). Fragments are software-pipelined:
// the k+4 loads are issued before the k-step WMMAs so the wave keeps a full
// load batch in flight instead of wait-0 before every WMMA.
// ---------------------------------------------------------------------------
__global__ __launch_bounds__(32)
void wmma_gemm_nt(const float* __restrict__ A, int lda,
                  const float* __restrict__ B, int ldb,
                  float* __restrict__ C, int ldc,
                  int K, const float* __restrict__ bias)
{
    const int lane = threadIdx.x;                 // wave32
    const int row0 = blockIdx.y * 16;
    const int col0 = blockIdx.x * 64;             // 4 x 16-wide column tiles
    const int m    = lane & 15;
    const int koff = (lane >> 4) << 1;            // 0 for lanes 0-15, 2 for 16-31

    const float* ap  = A + (size_t)(row0 + m) * lda + koff;
    const float* bp0 = B + (size_t)(col0 + m) * ldb + koff;
    const float* bp1 = bp0 + (size_t)16 * ldb;
    const float* bp2 = bp1 + (size_t)16 * ldb;
    const float* bp3 = bp2 + (size_t)16 * ldb;

    v8f c0 = {0.f,0.f,0.f,0.f,0.f,0.f,0.f,0.f};
    v8f c1 = c0, c2 = c0, c3 = c0;

    // prologue: fragments for k = 0
    v2f a  = *(const v2f*)(ap);
    v2f b0 = *(const v2f*)(bp0);
    v2f b1 = *(const v2f*)(bp1);
    v2f b2 = *(const v2f*)(bp2);
    v2f b3 = *(const v2f*)(bp3);

    for (int k = 4; k < K; k += 4) {
        // issue next-step loads before consuming current fragments
        v2f an  = *(const v2f*)(ap  + k);
        v2f bn0 = *(const v2f*)(bp0 + k);
        v2f bn1 = *(const v2f*)(bp1 + k);
        v2f bn2 = *(const v2f*)(bp2 + k);
        v2f bn3 = *(const v2f*)(bp3 + k);

        c0 = __builtin_amdgcn_wmma_f32_16x16x4_f32(false, a, false, b0,
                                                   (short)0, c0, false, false);
        c1 = __builtin_amdgcn_wmma_f32_16x16x4_f32(false, a, false, b1,
                                                   (short)0, c1, false, false);
        c2 = __builtin_amdgcn_wmma_f32_16x16x4_f32(false, a, false, b2,
                                                   (short)0, c2, false, false);
        c3 = __builtin_amdgcn_wmma_f32_16x16x4_f32(false, a, false, b3,
                                                   (short)0, c3, false, false);
        a = an; b0 = bn0; b1 = bn1; b2 = bn2; b3 = bn3;
    }
    // epilogue: last k-step
    c0 = __builtin_amdgcn_wmma_f32_16x16x4_f32(false, a, false, b0,
                                               (short)0, c0, false, false);
    c1 = __builtin_amdgcn_wmma_f32_16x16x4_f32(false, a, false, b1,
                                               (short)0, c1, false, false);
    c2 = __builtin_amdgcn_wmma_f32_16x16x4_f32(false, a, false, b2,
                                               (short)0, c2, false, false);
    c3 = __builtin_amdgcn_wmma_f32_16x16x4_f32(false, a, false, b3,
                                               (short)0, c3, false, false);

    const int rbase = row0 + (lane >> 4) * 8;     // lanes 16-31 hold M=8..15
    v8f acc[4] = {c0, c1, c2, c3};
#pragma unroll
    for (int t = 0; t < 4; ++t) {
        const int col  = col0 + t * 16 + m;
        const float badd = bias ? bias[col] : 0.0f;
#pragma unroll
        for (int r = 0; r < 8; ++r) {
            C[(size_t)(rbase + r) * ldc + col] = acc[t][r] + badd;
        }
    }
}

// ---------------------------------------------------------------------------
// Per-(node,head) attention logits: el = <feat[n,h,:], al[h,:]>, er likewise.
// t = n*H + h, and feat row offset n*256 + h*64 == t*64.
// ---------------------------------------------------------------------------
__global__ void attn_scores(const float* __restrict__ feat,
                            const float* __restrict__ al,
                            const float* __restrict__ ar,
                            float* __restrict__ el, float* __restrict__ er)
{
    int t = blockIdx.x * blockDim.x + threadIdx.x;
    if (t >= NN * HH) return;
    const int hh = t & 3;
    const float4* fp  = (const float4*)(feat + (size_t)t * 64);
    const float4* alp = (const float4*)(al + hh * 64);
    const float4* arp = (const float4*)(ar + hh * 64);
    float sl = 0.f, sr = 0.f;
#pragma unroll
    for (int i = 0; i < 16; ++i) {
        float4 f = fp[i], a = alp[i], r = arp[i];
        sl += f.x*a.x + f.y*a.y + f.z*a.z + f.w*a.w;
        sr += f.x*r.x + f.y*r.y + f.z*r.z + f.w*r.w;
    }
    el[t] = sl;
    er[t] = sr;
}

__device__ __forceinline__ float leaky(float x) { return x > 0.f ? x : NEG * x; }

// float atomic max via orderable int trick (portable, lowers to global atomics)
__device__ __forceinline__ void atomicMaxF(float* addr, float v)
{
    if (v >= 0.f) atomicMax((int*)addr, __float_as_int(v));
    else          atomicMin((unsigned int*)addr, __float_as_uint(v));
}

__global__ void init_sem(const float* __restrict__ b1, const float* __restrict__ b2,
                         float* __restrict__ sem)
{
    int t = blockIdx.x * blockDim.x + threadIdx.x;
    if (t >= NN * 2 * HD) return;
    int j = t & (2 * HD - 1);
    sem[t] = (j < HD) ? b1[j] : b2[j - HD];
}

__global__ void init_nh(float* __restrict__ emax, float* __restrict__ denom)
{
    int t = blockIdx.x * blockDim.x + threadIdx.x;
    if (t >= NN * HH) return;
    emax[t]  = -3.0e38f;
    denom[t] = 0.0f;
}

// Pass A: segment max of leaky(el[src]+er[dst]) over dst
__global__ void edge_max(const int* __restrict__ src, const int* __restrict__ dst,
                         const float* __restrict__ el, const float* __restrict__ er,
                         float* __restrict__ emax)
{
    int t = blockIdx.x * blockDim.x + threadIdx.x;
    if (t >= EE * HH) return;
    int e = t >> 2, hh = t & 3;
    int s = src[e], d = dst[e];
    float x = leaky(el[s * 4 + hh] + er[d * 4 + hh]);
    atomicMaxF(&emax[d * 4 + hh], x);
}

// Pass B: denom[dst] += exp(e - emax[dst])
__global__ void edge_denom(const int* __restrict__ src, const int* __restrict__ dst,
                           const float* __restrict__ el, const float* __restrict__ er,
                           const float* __restrict__ emax, float* __restrict__ denom)
{
    int t = blockIdx.x * blockDim.x + threadIdx.x;
    if (t >= EE * HH) return;
    int e = t >> 2, hh = t & 3;
    int s = src[e], d = dst[e];
    int di = d * 4 + hh;
    float x = leaky(el[s * 4 + hh] + er[d * 4 + hh]);
    atomicAdd(&denom[di], expf(x - emax[di]));
}

// Pass C: sem[dst, colOff + i] += feat[src, i] * alpha    (1 block per edge,
// 256 threads = 1 feature each; gather + atomic stream is 128B/wave coalesced
// and stays L2-resident: feat=51MB, sem=102MB < 192MB L2)
__global__ __launch_bounds__(256)
void edge_aggregate(const int* __restrict__ src, const int* __restrict__ dst,
                    const float* __restrict__ feat,
                    const float* __restrict__ el, const float* __restrict__ er,
                    const float* __restrict__ emax, const float* __restrict__ denom,
                    float* __restrict__ sem, int colOff)
{
    int e = blockIdx.x;
    int i = threadIdx.x;            // 0..255  (head = i>>6)
    int s = src[e], d = dst[e];
    int hh = i >> 6;
    int di = d * 4 + hh;
    float x = leaky(el[s * 4 + hh] + er[di]);
    float alpha = expf(x - emax[di]) / denom[di];
    atomicAdd(sem + (size_t)d * (2 * HD) + colOff + i,
              feat[(size_t)s * HD + i] * alpha);
}

// ---------------------------------------------------------------------------
extern "C" void kernel_launch(void* const* d_in, const int* in_sizes, int n_in,
                              void* d_out, int out_size, void* d_ws, size_t ws_size,
                              hipStream_t stream)
{
    const float* h   = (const float*)d_in[0];
    const float* Wg1 = (const float*)d_in[1];
    const float* al1 = (const float*)d_in[2];
    const float* ar1 = (const float*)d_in[3];
    const float* b1  = (const float*)d_in[4];
    const float* Wg2 = (const float*)d_in[5];
    const float* al2 = (const float*)d_in[6];
    const float* ar2 = (const float*)d_in[7];
    const float* b2  = (const float*)d_in[8];
    const float* Wfc = (const float*)d_in[9];
    const float* bfc = (const float*)d_in[10];
    const int* src1  = (const int*)d_in[11];
    const int* dst1  = (const int*)d_in[12];
    const int* src2  = (const int*)d_in[13];
    const int* dst2  = (const int*)d_in[14];
    float* out = (float*)d_out;

    // workspace layout (floats)
    float* ws    = (float*)d_ws;
    float* feat  = ws;                                    // N*HD      = 12.8M
    float* el    = feat  + (size_t)NN * HD;               // N*H
    float* er    = el    + (size_t)NN * HH;
    float* emax  = er    + (size_t)NN * HH;
    float* denom = emax  + (size_t)NN * HH;
    float* sem   = denom + (size_t)NN * HH;               // N*2HD     = 25.6M

    const int T256 = 256;
    // sem <- broadcast biases (scatter-adds accumulate on top)
    init_sem<<<(NN * 2 * HD + T256 - 1) / T256, T256, 0, stream>>>(b1, b2, sem);

    dim3 gemmGrid(HD / 64, NN / 16);          // 4 x 3125 stripes (16x64 per wave)

    const float* Wg[2]  = {Wg1, Wg2};
    const float* alw[2] = {al1, al2};
    const float* arw[2] = {ar1, ar2};
    const int*   srcs[2] = {src1, src2};
    const int*   dsts[2] = {dst1, dst2};

    for (int rel = 0; rel < 2; ++rel) {
        init_nh<<<(NN * HH + T256 - 1) / T256, T256, 0, stream>>>(emax, denom);

        // feat = h @ Wg^T   (fp32 WMMA)
        wmma_gemm_nt<<<gemmGrid, 32, 0, stream>>>(h, INS, Wg[rel], INS,
                                                  feat, HD, INS, nullptr);

        attn_scores<<<(NN * HH + T256 - 1) / T256, T256, 0, stream>>>(
            feat, alw[rel], arw[rel], el, er);

        edge_max<<<(EE * HH + T256 - 1) / T256, T256, 0, stream>>>(
            srcs[rel], dsts[rel], el, er, emax);

        edge_denom<<<(EE * HH + T256 - 1) / T256, T256, 0, stream>>>(
            srcs[rel], dsts[rel], el, er, emax, denom);

        edge_aggregate<<<EE, 256, 0, stream>>>(
            srcs[rel], dsts[rel], feat, el, er, emax, denom, sem, rel * HD);
    }

    // out = sem @ Wfc^T + bfc   (fp32 WMMA, K = 512)
    wmma_gemm_nt<<<gemmGrid, 32, 0, stream>>>(sem, 2 * HD, Wfc, 2 * HD,
                                              out, HD, 2 * HD, bfc);
}